// ClebschCombiningSingleUnrolled_27839978013283
// MI455X (gfx1250) — compile-verified
//
#include <hip/hip_runtime.h>
#include <hip/hip_bf16.h>
#include <stdint.h>

// ---------------------------------------------------------------------------
// Clebsch combine (L1=L2=3, LAMBD=2), reduced to a sparse bilinear form:
//   out[m] = sum_t W_t * a[p_t] * b[q_t]   (43 products, 36 weights)
// Memory-bound: 760 MB traffic -> ~33us floor @ 23.3 TB/s. Data path uses
// CDNA5 async global<->LDS DMA (ASYNCcnt) instead of VGPR staging.
// ---------------------------------------------------------------------------

#define BLOCK       256
#define TILE_PAIRS  1024                      // pairs per block
#define PP          (TILE_PAIRS / BLOCK)      // 4 pairs per thread
#define X_WORDS     (TILE_PAIRS * 7)          // 7168 floats per input tile
#define O_WORDS     (TILE_PAIRS * 5)          // 5120 floats per output tile
#define LDS_WORDS   (X_WORDS * 2 + O_WORDS)   // 19456 floats = 77,824 B

__device__ __forceinline__ unsigned lds_off(const void* p) {
    // LDS aperture: flat addr low 32 bits == LDS byte offset
    return (unsigned)(uintptr_t)p;
}

__device__ __forceinline__ void async_load_b128(unsigned lds_addr, const void* base, unsigned byte_off) {
    asm volatile("global_load_async_to_lds_b128 %0, %1, %2"
                 :: "v"(lds_addr), "v"(byte_off), "s"(base) : "memory");
}
__device__ __forceinline__ void async_load_b32(unsigned lds_addr, const void* base, unsigned byte_off) {
    asm volatile("global_load_async_to_lds_b32 %0, %1, %2"
                 :: "v"(lds_addr), "v"(byte_off), "s"(base) : "memory");
}
__device__ __forceinline__ void async_store_b128(void* base, unsigned byte_off, unsigned lds_addr) {
    asm volatile("global_store_async_from_lds_b128 %0, %1, %2"
                 :: "v"(byte_off), "v"(lds_addr), "s"(base) : "memory");
}
__device__ __forceinline__ void async_store_b32(void* base, unsigned byte_off, unsigned lds_addr) {
    asm volatile("global_store_async_from_lds_b32 %0, %1, %2"
                 :: "v"(byte_off), "v"(lds_addr), "s"(base) : "memory");
}
__device__ __forceinline__ void wait_async0() {
    asm volatile("s_wait_asynccnt 0x0" ::: "memory");
}

// --- Prologue: fold clebsch into 36 final weights ---------------------------
__global__ void cg_weights_kernel(const float* __restrict__ cg, float* __restrict__ W) {
    if (threadIdx.x != 0) return;
    const float h = 0.70710678118654752440f;   // 1/sqrt(2)
    const float c60 = cg[42], c06 = cg[6],  c51 = cg[36], c15 = cg[12];
    const float c42 = cg[30], c24 = cg[18], c33 = cg[24];
    const float c61 = cg[43], c52 = cg[37], c43 = cg[31], c34 = cg[25];
    const float c25 = cg[19], c16 = cg[13];
    const float c62 = cg[44], c53 = cg[38], c44 = cg[32], c35 = cg[26], c26 = cg[20];
    // out[2]  (mu=0 real)
    W[0]  = -0.5f * (c60 + c06);   // a0b0, a6b6
    W[1]  =  0.5f * (c51 + c15);   // a1b1, a5b5
    W[2]  = -0.5f * (c42 + c24);   // a2b2, a4b4
    W[3]  =  c33;                  // a3b3
    // out[3]  (-sqrt2 * mu=1 real)
    W[4]  =  h * c61;  W[5]  = -h * c52;  W[6]  = c43;  W[7] = c34;
    W[8]  = -h * c25;  W[9]  =  h * c16;
    // out[1]  (-sqrt2 * mu=1 imag)
    W[10] = -h * c61;  W[11] =  h * c61;  W[12] =  h * c52;  W[13] = -h * c52;
    W[14] =  c43;      W[15] =  c34;      W[16] = -h * c25;  W[17] =  h * c25;
    W[18] =  h * c16;  W[19] = -h * c16;
    // out[4]  (+sqrt2 * mu=2 real)
    W[20] = -h * c62;  W[21] = -h * c62;  W[22] = c53;       W[23] =  h * c44;
    W[24] = -h * c44;  W[25] =  c35;      W[26] = -h * c26;  W[27] = -h * c26;
    // out[0]  (+sqrt2 * mu=2 imag)
    W[28] =  h * c62;  W[29] = -h * c62;  W[30] = c53;       W[31] =  h * c44;
    W[32] =  h * c44;  W[33] =  c35;      W[34] = -h * c26;  W[35] =  h * c26;
}

// --- Main streaming kernel --------------------------------------------------
__global__ __launch_bounds__(BLOCK) void cg_combine_kernel(
        const float* __restrict__ X1, const float* __restrict__ X2,
        const float* __restrict__ W,  float* __restrict__ out, long long NF) {
    __shared__ float lds[LDS_WORDS];
    float* As = lds;
    float* Bs = lds + X_WORDS;
    float* Os = lds + 2 * X_WORDS;

    const int tid = threadIdx.x;
    const long long base = (long long)blockIdx.x * TILE_PAIRS;
    long long rem = NF - base;
    const int valid = (rem < (long long)TILE_PAIRS) ? (int)rem : TILE_PAIRS;

    const unsigned ldsA = lds_off(As);
    const unsigned ldsB = lds_off(Bs);
    const unsigned ldsO = lds_off(Os);
    const float* gA = X1 + base * 7;
    const float* gB = X2 + base * 7;
    float*       gO = out + base * 5;

    // ---- async global -> LDS tile load (ASYNCcnt path) ----
    if (valid == TILE_PAIRS) {
        // 28,672 B per input = 7 * (256 threads * 16 B)
#pragma unroll
        for (int c = 0; c < 7; ++c) {
            const unsigned off = (unsigned)(c * BLOCK + tid) * 16u;
            async_load_b128(ldsA + off, gA, off);
            async_load_b128(ldsB + off, gB, off);
        }
    } else {
        const int words = valid * 7;
        for (int w = tid; w < words; w += BLOCK) {
            const unsigned off = (unsigned)w * 4u;
            async_load_b32(ldsA + off, gA, off);
            async_load_b32(ldsB + off, gB, off);
        }
    }
    wait_async0();
    __syncthreads();

    // uniform scalar weight loads
    float w[36];
#pragma unroll
    for (int i = 0; i < 36; ++i) w[i] = W[i];

    // ---- compute: 43 products per pair (stride 7/5 -> conflict-free banks) --
#pragma unroll
    for (int it = 0; it < PP; ++it) {
        const int p = tid + it * BLOCK;
        if (p < valid) {
            const float* a = &As[p * 7];
            const float* b = &Bs[p * 7];
            const float a0=a[0], a1=a[1], a2=a[2], a3=a[3], a4=a[4], a5=a[5], a6=a[6];
            const float b0=b[0], b1=b[1], b2=b[2], b3=b[3], b4=b[4], b5=b[5], b6=b[6];

            const float o2 = w[0]*(a0*b0 + a6*b6) + w[1]*(a1*b1 + a5*b5)
                           + w[2]*(a2*b2 + a4*b4) + w[3]*(a3*b3);
            const float o3 = w[4]*(a6*b5 + a0*b1) + w[5]*(a5*b4 + a1*b2)
                           + w[6]*(a4*b3) + w[7]*(a3*b4)
                           + w[8]*(a4*b5 + a2*b1) + w[9]*(a5*b6 + a1*b0);
            const float o1 = w[10]*(a6*b1) + w[11]*(a0*b5) + w[12]*(a5*b2)
                           + w[13]*(a1*b4) + w[14]*(a2*b3) + w[15]*(a3*b2)
                           + w[16]*(a4*b1) + w[17]*(a2*b5) + w[18]*(a5*b0)
                           + w[19]*(a1*b6);
            const float o4 = w[20]*(a6*b4) + w[21]*(a0*b2) + w[22]*(a5*b3)
                           + w[23]*(a4*b4) + w[24]*(a2*b2) + w[25]*(a3*b5)
                           + w[26]*(a4*b6) + w[27]*(a2*b0);
            const float o0 = w[28]*(a6*b2) + w[29]*(a0*b4) + w[30]*(a1*b3)
                           + w[31]*(a4*b2) + w[32]*(a2*b4) + w[33]*(a3*b1)
                           + w[34]*(a4*b0) + w[35]*(a2*b6);

            float* o = &Os[p * 5];
            o[0] = o0; o[1] = o1; o[2] = o2; o[3] = o3; o[4] = o4;
        }
    }
    __syncthreads();

    // ---- async LDS -> global tile store ----
    if (valid == TILE_PAIRS) {
        // 20,480 B = 5 * (256 threads * 16 B)
#pragma unroll
        for (int c = 0; c < 5; ++c) {
            const unsigned off = (unsigned)(c * BLOCK + tid) * 16u;
            async_store_b128(gO, off, ldsO + off);
        }
    } else {
        const int words = valid * 5;
        for (int wd = tid; wd < words; wd += BLOCK) {
            const unsigned off = (unsigned)wd * 4u;
            async_store_b32(gO, off, ldsO + off);
        }
    }
    // S_ENDPGM performs an implicit S_WAIT_IDLE (covers ASYNCcnt).
}

extern "C" void kernel_launch(void* const* d_in, const int* in_sizes, int n_in,
                              void* d_out, int out_size, void* d_ws, size_t ws_size,
                              hipStream_t stream) {
    const float* X1 = (const float*)d_in[0];
    const float* X2 = (const float*)d_in[1];
    const float* CG = (const float*)d_in[2];
    float* out = (float*)d_out;
    float* W   = (float*)d_ws;   // 36 floats of folded weights

    const long long NF = (long long)in_sizes[0] / 7;   // N*F pairs
    cg_weights_kernel<<<1, 32, 0, stream>>>(CG, W);
    const int grid = (int)((NF + TILE_PAIRS - 1) / TILE_PAIRS);
    cg_combine_kernel<<<grid, BLOCK, 0, stream>>>(X1, X2, W, out, NF);
}